// SLODEWrapper_70265664963014
// MI455X (gfx1250) — compile-verified
//
#include <hip/hip_runtime.h>

typedef __attribute__((ext_vector_type(2))) float v2f;
typedef __attribute__((ext_vector_type(8))) float v8f;

#define NX 128
#define NU 64
#define NY 64
#define BATCH 32
#define TT 4096
#define NCHUNK 32
#define CLEN 128
#define XPITCH 132   // 128 + 4 pad -> conflict-free strided LDS reads
#define UPITCH 68    // 64 + 4 pad

// D = A(16x4,f32) * B(4x16,f32) + C(16x16,f32)  — native fp32 WMMA on CDNA5
__device__ __forceinline__ v8f wmma_f32(v2f a, v2f b, v8f c) {
  return __builtin_amdgcn_wmma_f32_16x16x4_f32(false, a, false, b, (short)0, c,
                                               false, false);
}

// ---------------------------------------------------------------------------
// Kernel 1: E = 0.5*(S11 + S22) + tol*I ; F = S21   with S = M M^T
// ---------------------------------------------------------------------------
__global__ void k_EF(const float* __restrict__ M, float* __restrict__ E,
                     float* __restrict__ F) {
  const int i = blockIdx.x;
  const int j = threadIdx.x;
  const float* Mi  = M + (size_t)i * 256;
  const float* Min = M + (size_t)(i + NX) * 256;
  const float* Mj  = M + (size_t)j * 256;
  const float* Mjn = M + (size_t)(j + NX) * 256;
  float d11 = 0.f, d22 = 0.f, d21 = 0.f;
  for (int q = 0; q < 256; ++q) {
    const float a = Mi[q], an = Min[q], b = Mj[q], bn = Mjn[q];
    d11 += a * b;
    d22 += an * bn;
    d21 += an * b;
  }
  float e = 0.5f * (d11 + d22);
  if (i == j) e += 1e-9f;
  E[i * NX + j] = e;
  F[i * NX + j] = d21;
}

// ---------------------------------------------------------------------------
// Kernel 2: solve E * A = F, A overwrites F.  E is SPD -> GE w/o pivoting.
// ---------------------------------------------------------------------------
__global__ void k_solve(float* __restrict__ E, float* __restrict__ F) {
  const int tid = threadIdx.x;
  for (int p = 0; p < NX - 1; ++p) {
    __syncthreads();
    const float inv = 1.0f / E[p * NX + p];
    const int i = p + 1 + tid;
    if (i < NX) {
      const float f = E[i * NX + p] * inv;
      for (int q = p; q < NX; ++q) E[i * NX + q] -= f * E[p * NX + q];
      for (int q = 0; q < NX; ++q) F[i * NX + q] -= f * F[p * NX + q];
    }
  }
  for (int p = NX - 1; p >= 0; --p) {
    __syncthreads();
    if (tid < NX) {
      float s = F[p * NX + tid];
      for (int q = p + 1; q < NX; ++q) s -= E[p * NX + q] * F[q * NX + tid];
      F[p * NX + tid] = s / E[p * NX + p];
    }
  }
}

// ---------------------------------------------------------------------------
// Kernel 3: AL = A^128 via 7 squarings (ping-pong T0 <-> AL, ends in AL)
// ---------------------------------------------------------------------------
__global__ void k_power(const float* __restrict__ A, float* __restrict__ T0,
                        float* __restrict__ AL) {
  const int tid = threadIdx.x;
  for (int e = tid; e < NX * NX; e += 1024) T0[e] = A[e];
  __syncthreads();
  float* cur = T0;
  float* nxt = AL;
  for (int it = 0; it < 7; ++it) {
    for (int e = tid; e < NX * NX; e += 1024) {
      const int i = e >> 7, j = e & 127;
      float s = 0.f;
      for (int k = 0; k < NX; ++k) s += cur[i * NX + k] * cur[k * NX + j];
      nxt[e] = s;
    }
    __syncthreads();
    float* t = cur; cur = nxt; nxt = t;
  }
  // after 7 iterations result lives in AL
}

// ---------------------------------------------------------------------------
// Kernel 4/6: per-chunk scan.  emit_y==0: local scan (x0=0), store chunk-end
// state to zend.  emit_y==1: start from xstart[c], emit y each step.
// A and B_mat operand slices resident in VGPRs; C operands in LDS
// (transposed, conflict-free) to stay under the 256-VGPR ceiling.
// ---------------------------------------------------------------------------
__global__ void __launch_bounds__(512, 1)
k_scan(const float* __restrict__ u, const float* __restrict__ A,
       const float* __restrict__ Bm, const float* __restrict__ Cm,
       const float* __restrict__ xstart, float* __restrict__ zend,
       float* __restrict__ out, int emit_y) {
  __shared__ float xbuf[BATCH * XPITCH];   // x state   [m][k], pitch 132
  __shared__ float ubuf[BATCH * UPITCH];   // u_t stage [b][k], pitch 68
  __shared__ float cbufT[NX * NY];         // C^T       [k][n], pitch 64

  const int tid  = threadIdx.x;
  const int wave = tid >> 5;
  const int lane = tid & 31;
  const int mi = wave & 1;
  const int ni = wave >> 1;
  const int lr = lane & 15;
  const int kh = (lane >> 4) << 1;  // B-operand K sub-row: 0 or 2
  const int c  = blockIdx.x;

  // Resident B-operands: element (k,n) of P^T read as P[n][k].
  v2f Aop[32];
  v2f Bop[16];
  {
    const int n = ni * 16 + lr;
#pragma unroll
    for (int kk = 0; kk < 32; ++kk) {
      const int k = kk * 4 + kh;
      Aop[kk].x = A[n * NX + k];
      Aop[kk].y = A[n * NX + k + 1];
    }
#pragma unroll
    for (int kk = 0; kk < 16; ++kk) {
      const int k = kk * 4 + kh;
      Bop[kk].x = Bm[n * NU + k];
      Bop[kk].y = Bm[n * NU + k + 1];
    }
  }

  // stage C^T into LDS (one-time; conflict-free LDS writes)
  if (emit_y) {
    for (int e = tid; e < NY * NX; e += 512) {
      const int n = e & 63, k = e >> 6;
      cbufT[k * NY + n] = Cm[n * NX + k];
    }
  }
  // init state
  for (int e = tid; e < BATCH * NX; e += 512) {
    const int b = e >> 7, k = e & 127;
    xbuf[b * XPITCH + k] = emit_y ? xstart[(c * BATCH + b) * NX + k] : 0.0f;
  }
  // stage u for s=0 (coalesced)
  {
    const int t = c * CLEN;
    for (int e = tid; e < BATCH * NU; e += 512) {
      const int b = e >> 6, k = e & 63;
      ubuf[b * UPITCH + k] = u[((size_t)b * TT + t) * NU + k];
    }
  }
  __syncthreads();

  const int arow = (mi * 16 + lr) * XPITCH;
  const int urow = (mi * 16 + lr) * UPITCH;
  const int mrow = mi * 16 + (lane >> 4) * 8;  // base M row of acc tile
  const int ncol = ni * 16 + lr;               // this wave/lane N column

  for (int s = 0; s < CLEN; ++s) {
    const int t = c * CLEN + s;

    // y_t = x_t @ C^T  (waves ni<4 cover the 32x64 output)
    if (emit_y && ni < 4) {
      v8f y = {};
#pragma unroll
      for (int kk = 0; kk < 32; ++kk) {
        const int k = kk * 4 + kh;
        v2f a, cb;
        a.x  = xbuf[arow + k];
        a.y  = xbuf[arow + k + 1];
        cb.x = cbufT[k * NY + ncol];
        cb.y = cbufT[(k + 1) * NY + ncol];
        y = wmma_f32(a, cb, y);
      }
#pragma unroll
      for (int r = 0; r < 8; ++r) {
        const int b = mrow + r;
        out[((size_t)b * TT + t) * NY + ncol] = y[r];
      }
    }

    // x_{t+1} = u_t @ B^T + x_t @ A^T
    v8f acc = {};
#pragma unroll
    for (int kk = 0; kk < 16; ++kk) {
      const int k = kk * 4 + kh;
      v2f a;
      a.x = ubuf[urow + k];
      a.y = ubuf[urow + k + 1];
      acc = wmma_f32(a, Bop[kk], acc);
    }
#pragma unroll
    for (int kk = 0; kk < 32; ++kk) {
      const int k = kk * 4 + kh;
      v2f a;
      a.x = xbuf[arow + k];
      a.y = xbuf[arow + k + 1];
      acc = wmma_f32(a, Aop[kk], acc);
    }
    __syncthreads();  // done reading xbuf/ubuf
#pragma unroll
    for (int r = 0; r < 8; ++r)
      xbuf[(mrow + r) * XPITCH + ncol] = acc[r];
    if (s + 1 < CLEN) {
      const int tn = t + 1;
      for (int e = tid; e < BATCH * NU; e += 512) {
        const int b = e >> 6, k = e & 63;
        ubuf[b * UPITCH + k] = u[((size_t)b * TT + tn) * NU + k];
      }
    }
    __syncthreads();
  }

  if (!emit_y) {
    for (int e = tid; e < BATCH * NX; e += 512) {
      const int b = e >> 7, k = e & 127;
      zend[(c * BATCH + b) * NX + k] = xbuf[b * XPITCH + k];
    }
  }
}

// ---------------------------------------------------------------------------
// Kernel 5: chunk-boundary fix-up.
// xstart[0]=x0 ; xstart[c+1] = xstart[c] @ (A^128)^T + zend[c]
// ---------------------------------------------------------------------------
__global__ void __launch_bounds__(512, 1)
k_phase2(const float* __restrict__ AL, const float* __restrict__ x0,
         const float* __restrict__ zend, float* __restrict__ xstart) {
  __shared__ float xbuf[BATCH * XPITCH];
  const int tid  = threadIdx.x;
  const int wave = tid >> 5;
  const int lane = tid & 31;
  const int mi = wave & 1;
  const int ni = wave >> 1;
  const int lr = lane & 15;
  const int kh = (lane >> 4) << 1;

  v2f Lop[32];
  {
    const int n = ni * 16 + lr;
#pragma unroll
    for (int kk = 0; kk < 32; ++kk) {
      const int k = kk * 4 + kh;
      Lop[kk].x = AL[n * NX + k];
      Lop[kk].y = AL[n * NX + k + 1];
    }
  }
  for (int e = tid; e < BATCH * NX; e += 512)
    xbuf[(e >> 7) * XPITCH + (e & 127)] = x0[e & 127];
  __syncthreads();

  const int arow = (mi * 16 + lr) * XPITCH;
  const int mrow = mi * 16 + (lane >> 4) * 8;
  const int ncol = ni * 16 + lr;

  for (int cc = 0; cc < NCHUNK; ++cc) {
    for (int e = tid; e < BATCH * NX; e += 512)
      xstart[(cc * BATCH + (e >> 7)) * NX + (e & 127)] =
          xbuf[(e >> 7) * XPITCH + (e & 127)];
    v8f acc;
#pragma unroll
    for (int r = 0; r < 8; ++r)
      acc[r] = zend[(cc * BATCH + mrow + r) * NX + ncol];
#pragma unroll
    for (int kk = 0; kk < 32; ++kk) {
      const int k = kk * 4 + kh;
      v2f a;
      a.x = xbuf[arow + k];
      a.y = xbuf[arow + k + 1];
      acc = wmma_f32(a, Lop[kk], acc);
    }
    __syncthreads();
#pragma unroll
    for (int r = 0; r < 8; ++r)
      xbuf[(mrow + r) * XPITCH + ncol] = acc[r];
    __syncthreads();
  }
}

// ---------------------------------------------------------------------------
extern "C" void kernel_launch(void* const* d_in, const int* in_sizes, int n_in,
                              void* d_out, int out_size, void* d_ws,
                              size_t ws_size, hipStream_t stream) {
  const float* u  = (const float*)d_in[0];   // (32,4096,64)
  const float* M  = (const float*)d_in[1];   // (256,256)
  const float* Bm = (const float*)d_in[2];   // (128,64)
  const float* Cm = (const float*)d_in[3];   // (64,128)
  const float* x0 = (const float*)d_in[4];   // (128,)
  float* out = (float*)d_out;                // (32,4096,64)

  float* ws  = (float*)d_ws;
  float* E   = ws;              // 16384
  float* A   = ws + 16384;      // 16384 (Fm, overwritten by A)
  float* AL  = ws + 32768;      // 16384 (A^128)
  float* TMP = ws + 49152;      // 16384
  float* ZE  = ws + 65536;      // 32768 chunk-end local states
  float* XS  = ws + 98304;      // 32768 chunk-start states

  k_EF<<<128, 128, 0, stream>>>(M, E, A);
  k_solve<<<1, 128, 0, stream>>>(E, A);
  k_power<<<1, 1024, 0, stream>>>(A, TMP, AL);
  k_scan<<<NCHUNK, 512, 0, stream>>>(u, A, Bm, Cm, nullptr, ZE, nullptr, 0);
  k_phase2<<<1, 512, 0, stream>>>(AL, x0, ZE, XS);
  k_scan<<<NCHUNK, 512, 0, stream>>>(u, A, Bm, Cm, XS, ZE, out, 1);
}